// MessageAggregation_36730560315453
// MI455X (gfx1250) — compile-verified
//
#include <hip/hip_runtime.h>

// ---------------------------------------------------------------------------
// MI455X (gfx1250) implementation.
//
// Phase 1 (dominant cost, ~13.5us at 23.3 TB/s): stream 307MB of messages once
//   (non-temporal float4 loads), scale by edge_cutoff, scatter with native f32
//   global atomics into a 19.2MB accumulator that lives in the 192MB L2.
// Phase 2 (~1.7us): block-diagonal equivariant linear done with
//   V_WMMA_F32_16X16X4_F32 (exact f32 math, matches the reference), one wave32
//   per 16-node tile, K accumulated in steps of 4.  Full tiles (the only kind
//   for n_nodes=20000) take a branch-free store path with immediate offsets.
// ---------------------------------------------------------------------------

typedef __attribute__((ext_vector_type(2))) float v2f;
typedef __attribute__((ext_vector_type(4))) float v4f;
typedef __attribute__((ext_vector_type(8))) float v8f;

#define IRREPS_DIM 240   // 64*1 + 32*3 + 16*5

// ---------------------------------------------------------------------------
// Phase 1: weighted scatter-add.  One thread per (edge, float4-chunk).
// ---------------------------------------------------------------------------
__global__ void __launch_bounds__(256)
scatter_add_kernel(const v4f* __restrict__ msg4,
                   const long long* __restrict__ edge_index,
                   const float* __restrict__ cutoff,
                   float* __restrict__ agg,
                   int n_edges)
{
    const int chunks = IRREPS_DIM / 4;              // 60 float4 per edge
    int gid = blockIdx.x * blockDim.x + threadIdx.x;
    if (gid >= n_edges * chunks) return;

    int edge  = gid / chunks;                       // magic-mul division
    int chunk = gid - edge * chunks;

    // messages are read exactly once -> non-temporal, keep agg resident in L2
    v4f m   = __builtin_nontemporal_load(&msg4[gid]);
    float w = cutoff[edge];
    int  t  = (int)edge_index[2 * edge + 1];        // target node (int64 input)

    float* dst = agg + (size_t)t * IRREPS_DIM + chunk * 4;
    // native GLOBAL_ATOMIC_ADD_F32 (no CAS loop)
    unsafeAtomicAdd(dst + 0, m.x * w);
    unsafeAtomicAdd(dst + 1, m.y * w);
    unsafeAtomicAdd(dst + 2, m.z * w);
    unsafeAtomicAdd(dst + 3, m.w * w);
}

// ---------------------------------------------------------------------------
// Phase 2: equivariant linear with V_WMMA_F32_16X16X4_F32.
//
// One 16x16 output tile:  D[m][n] = sum_i x[node m, i] * W[o_base+n, i]
// A (16x4, f32, 2 VGPR): lanes 0-15 hold K=0(v0),1(v1); lanes 16-31 K=2,3.
// B (4x16, f32, 2 VGPR): row-striped: lanes 0-15 K=0(v0),1(v1); 16-31 K=2,3.
// C/D (16x16 f32, 8 VGPR): VGPR j = rows M=j (lanes 0-15) and M=j+8 (16-31).
// ---------------------------------------------------------------------------
__device__ __forceinline__ void irrep_tile(const float* __restrict__ x,
                                           const float* __restrict__ W,
                                           float* __restrict__ out,
                                           int base, int n_nodes, bool full,
                                           int mul, int deg, int off,
                                           int d, int o_base, float scale,
                                           int lane)
{
    const int row = lane & 15;     // A row / B-D column within the tile
    const int kh  = lane >> 4;     // K half-select

    const int nodeA = full ? (base + row) : min(base + row, n_nodes - 1);
    const float* xrow = x + (size_t)nodeA * IRREPS_DIM + off + d;   // stride deg
    const float* wrow = W + (o_base + row) * mul;                   // row-major W[o][i]

    v8f c = {0.f, 0.f, 0.f, 0.f, 0.f, 0.f, 0.f, 0.f};
    for (int ib = 0; ib < mul; ib += 4) {
        const int i0 = ib + kh * 2;
        v2f a, b;
        a.x = xrow[(i0 + 0) * deg];
        a.y = xrow[(i0 + 1) * deg];
        b.x = wrow[i0 + 0];
        b.y = wrow[i0 + 1];
        // D = A(16x4) * B(4x16) + C, exact f32 (RNE) -> v_wmma_f32_16x16x4_f32
        c = __builtin_amdgcn_wmma_f32_16x16x4_f32(
                /*neg_a=*/false, a, /*neg_b=*/false, b,
                /*c_mod=*/(short)0, c, /*reuse_a=*/false, /*reuse_b=*/false);
    }

    // D element (m=j+kh*8, n=row) -> out[node, off + (o_base+n)*deg + d]
    if (full) {
        // wave-uniform fast path: one base pointer per lane, constant byte
        // offsets j*240*4=j*960 (< 24-bit imm) -> branch-free 8-store clause
        float* op = out + (size_t)(base + kh * 8) * IRREPS_DIM
                        + off + (o_base + row) * deg + d;
#pragma unroll
        for (int j = 0; j < 8; ++j)
            op[(size_t)j * IRREPS_DIM] = c[j] * scale;
    } else {
#pragma unroll
        for (int j = 0; j < 8; ++j) {
            const int node = base + j + kh * 8;
            if (node < n_nodes)
                out[(size_t)node * IRREPS_DIM + off + (o_base + row) * deg + d] =
                    c[j] * scale;
        }
    }
}

__global__ void __launch_bounds__(256)
equivariant_linear_wmma(const float* __restrict__ agg,
                        const float* __restrict__ W0,
                        const float* __restrict__ W1,
                        const float* __restrict__ W2,
                        float* __restrict__ out,
                        int n_nodes, int n_tiles)
{
    const int lane = threadIdx.x & 31;              // wave32
    const int wave = threadIdx.x >> 5;
    const int tile = blockIdx.x * (blockDim.x >> 5) + wave;
    if (tile >= n_tiles) return;                    // wave-uniform exit, EXEC stays full
    const int base = tile * 16;
    const bool full = (base + 16 <= n_nodes);       // wave-uniform (always true here)

    // irrep 0: 64x0e, deg=1, offset 0, scale 1/sqrt(64)
    for (int ob = 0; ob < 64; ob += 16)
        irrep_tile(agg, W0, out, base, n_nodes, full, 64, 1, 0,   0, ob, 0.125f, lane);

    // irrep 1: 32x1o, deg=3, offset 64, scale 1/sqrt(32)
    for (int d = 0; d < 3; ++d)
        for (int ob = 0; ob < 32; ob += 16)
            irrep_tile(agg, W1, out, base, n_nodes, full, 32, 3, 64,  d, ob,
                       0.17677669529663687f, lane);

    // irrep 2: 16x2e, deg=5, offset 160, scale 1/sqrt(16)
    for (int d = 0; d < 5; ++d)
        irrep_tile(agg, W2, out, base, n_nodes, full, 16, 5, 160, d, 0, 0.25f, lane);
}

// ---------------------------------------------------------------------------
extern "C" void kernel_launch(void* const* d_in, const int* in_sizes, int n_in,
                              void* d_out, int out_size, void* d_ws, size_t ws_size,
                              hipStream_t stream)
{
    const v4f*       messages   = (const v4f*)d_in[0];       // (E, 240) f32
    const long long* edge_index = (const long long*)d_in[1]; // (E, 2) int64
    const float*     cutoff     = (const float*)d_in[2];     // (E,)
    const float*     W0         = (const float*)d_in[3];     // (64,64)
    const float*     W1         = (const float*)d_in[4];     // (32,32)
    const float*     W2         = (const float*)d_in[5];     // (16,16)
    float*           out        = (float*)d_out;

    const int n_edges = in_sizes[2];                 // edge_cutoff length
    const int n_nodes = out_size / IRREPS_DIM;       // n_nodes scalar lives on device

    float* agg = (float*)d_ws;
    const size_t agg_bytes = (size_t)n_nodes * IRREPS_DIM * sizeof(float);
    hipMemsetAsync(agg, 0, agg_bytes, stream);       // graph-capturable

    // Phase 1: scatter-add
    {
        const int total   = n_edges * (IRREPS_DIM / 4);
        const int threads = 256;
        const int blocks  = (total + threads - 1) / threads;
        scatter_add_kernel<<<blocks, threads, 0, stream>>>(
            messages, edge_index, cutoff, agg, n_edges);
    }

    // Phase 2: WMMA equivariant linear (1 wave per 16-node tile)
    {
        const int n_tiles = (n_nodes + 15) / 16;     // 1250 for 20000 nodes
        const int threads = 256;                     // 8 waves / block
        const int waves   = threads / 32;
        const int blocks  = (n_tiles + waves - 1) / waves;
        equivariant_linear_wmma<<<blocks, threads, 0, stream>>>(
            agg, W0, W1, W2, out, n_nodes, n_tiles);
    }
}